// simple_block_49435073577385
// MI455X (gfx1250) — compile-verified
//
#include <hip/hip_runtime.h>

#define NPTS 65536
#define NNEI 32
#define NK   15
#define INF  64
#define OUTF 128

// pitches (in floats) chosen for LDS bank-conflict avoidance (64 banks)
#define WBP 65   // weighted buffer pitch: 65*x % 64 == x -> lanes hit distinct banks
#define NXP 68   // neighbor-feature pitch: 272B rows, 16B-aligned for async B128
#define WLP 34   // influence pitch: 34*lm mod 64 distinct for lm in 0..15

#define NX_BASE (16 * 16 * WBP)                    // float offset of nx region in smem
#define WL_BASE (NX_BASE + 4 * NNEI * NXP)         // float offset of wl region

typedef float v2f __attribute__((ext_vector_type(2)));
typedef float v8f __attribute__((ext_vector_type(8)));

__constant__ static const float kINV_KPE = 1.0f / (1.2f * 0.1f / 5.0f); // 1/KP_EXTENT

__global__ __launch_bounds__(128) void kpconv_wmma(
    const float* __restrict__ qp,   // [NPTS][3]
    const float* __restrict__ sp,   // [NPTS][3]
    const int*   __restrict__ nbr,  // [NPTS][NNEI]
    const float* __restrict__ feat, // [NPTS][INF]
    const float* __restrict__ kp,   // [NK][3]
    const float* __restrict__ Wg,   // [NK][INF][OUTF] == [960][128]
    float*       __restrict__ out)  // [NPTS][OUTF] (raw pre-BN)
{
    extern __shared__ float smem[];
    const int tid  = threadIdx.x;
    const int wave = tid >> 5;
    const int lane = tid & 31;
    const int half = lane >> 4;   // which 16-lane half
    const int lm   = lane & 15;

    float* wbuf = smem;                                   // [16k][16p][WBP]
    float* nx   = smem + NX_BASE + wave * (NNEI * NXP);   // per-wave [32m][NXP]
    float* wl   = smem + WL_BASE + wave * (16 * WLP);     // per-wave [16k][WLP]

    const int pbase = blockIdx.x * 16;

    // ---- stage A+B: per wave, 4 points: influences + async gather + (w @ nx) WMMA ----
    for (int j = 0; j < 4; ++j) {
        const int pl = wave * 4 + j;        // local point 0..15
        const int p  = pbase + pl;

        // lane == neighbor index m
        const int m   = lane;
        const int idx = nbr[p * NNEI + m];

        // --- async DMA of this neighbor's 64 features straight into LDS ---
        // LDS byte address (relative to workgroup LDS base; only dynamic LDS here)
        const unsigned nx_lds = (unsigned)((NX_BASE + wave * (NNEI * NXP) + m * NXP)
                                           * sizeof(float));
        const float* gsrc = feat + (size_t)idx * INF;
        #pragma unroll
        for (int t = 0; t < INF / 4; ++t) {   // 16 x B128 = 256B per lane
            asm volatile("global_load_async_to_lds_b128 %0, %1, off offset:%2"
                         :: "v"(nx_lds), "v"(gsrc), "i"(t * 16)
                         : "memory");
        }

        // --- influence weights while the DMA is in flight ---
        const float qx = qp[p * 3 + 0], qy = qp[p * 3 + 1], qz = qp[p * 3 + 2];
        const float dx = sp[idx * 3 + 0] - qx;
        const float dy = sp[idx * 3 + 1] - qy;
        const float dz = sp[idx * 3 + 2] - qz;
        #pragma unroll
        for (int k = 0; k < NK; ++k) {
            const float ax = dx - kp[k * 3 + 0];
            const float ay = dy - kp[k * 3 + 1];
            const float az = dz - kp[k * 3 + 2];
            const float r  = sqrtf(ax * ax + ay * ay + az * az);
            wl[k * WLP + m] = fmaxf(0.0f, 1.0f - r * kINV_KPE);
        }
        wl[15 * WLP + m] = 0.0f;  // pad row of the 16x32 A tile

        // wait for the async LDS writes, fence the compiler
        asm volatile("s_wait_asynccnt 0x0" ::: "memory");
        __builtin_amdgcn_wave_barrier();

        // weighted[16k x 64i] = wl[16x32] @ nx[32x64] with f32 WMMA, K-steps of 4
        #pragma unroll
        for (int nt = 0; nt < 4; ++nt) {
            v8f c = {};
            #pragma unroll
            for (int kk = 0; kk < NNEI; kk += 4) {
                const int kb = kk + 2 * half;       // ISA A/B fragment: K split by lane-half
                v2f a, b;
                a[0] = wl[lm * WLP + kb];
                a[1] = wl[lm * WLP + kb + 1];
                b[0] = nx[kb * NXP + nt * 16 + lm];
                b[1] = nx[(kb + 1) * NXP + nt * 16 + lm];
                c = __builtin_amdgcn_wmma_f32_16x16x4_f32(
                        false, a, false, b, (short)0, c, false, false);
            }
            // C layout: VGPR r -> row (r + 8*half), col = lm
            #pragma unroll
            for (int r = 0; r < 8; ++r) {
                const int krow = r + 8 * half;       // k index (15 == pad, never read)
                wbuf[(krow * 16 + pl) * WBP + nt * 16 + lm] = c[r];
            }
        }
        __builtin_amdgcn_wave_barrier();
    }
    __syncthreads();

    // ---- stage C: out[16p x 128o] = A[16 x 960] @ Wflat[960 x 128] ----
    // each wave owns 2 of the 8 output 16-wide tiles
    for (int ot = wave * 2; ot < wave * 2 + 2; ++ot) {
        const int o0 = ot * 16;
        v8f c = {};
        #pragma unroll 4
        for (int kg = 0; kg < NK * INF; kg += 4) {
            const int kd = kg + 2 * half;   // kd..kd+1 stay within one k-slab (64 | steps)
            const int kb = kd >> 6;         // kernel-point index
            const int ib = kd & 63;         // input-feature index
            v2f a, b;
            a[0] = wbuf[(kb * 16 + lm) * WBP + ib];
            a[1] = wbuf[(kb * 16 + lm) * WBP + ib + 1];
            b[0] = Wg[(size_t)kd * OUTF + o0 + lm];
            b[1] = Wg[(size_t)(kd + 1) * OUTF + o0 + lm];
            c = __builtin_amdgcn_wmma_f32_16x16x4_f32(
                    false, a, false, b, (short)0, c, false, false);
        }
        #pragma unroll
        for (int r = 0; r < 8; ++r) {
            const int prow = r + 8 * half;   // point index within tile
            out[(size_t)(pbase + prow) * OUTF + o0 + lm] = c[r];
        }
    }
}

// ---- BN stats: deterministic two-level reduction (no float atomics) ----
__global__ __launch_bounds__(128) void stats_partial(const float* __restrict__ x,
                                                     float* __restrict__ part)
{
    const int f = threadIdx.x;        // feature 0..127
    const int b = blockIdx.x;         // 64 blocks, 1024 rows each
    float s = 0.0f, s2 = 0.0f;
    const int base = b * 1024;
    for (int r = 0; r < 1024; ++r) {
        const float v = x[(size_t)(base + r) * OUTF + f];
        s += v;
        s2 += v * v;
    }
    part[b * OUTF + f]        = s;
    part[(64 + b) * OUTF + f] = s2;
}

__global__ __launch_bounds__(128) void stats_final(const float* __restrict__ part,
                                                   float* __restrict__ stats)
{
    const int f = threadIdx.x;
    float s = 0.0f, s2 = 0.0f;
    for (int b = 0; b < 64; ++b) {
        s  += part[b * OUTF + f];
        s2 += part[(64 + b) * OUTF + f];
    }
    stats[f]        = s;
    stats[OUTF + f] = s2;
}

__global__ __launch_bounds__(256) void bn_leaky(float* __restrict__ x,
                                                const float* __restrict__ stats,
                                                const float* __restrict__ gamma,
                                                const float* __restrict__ beta)
{
    const size_t idx = (size_t)blockIdx.x * blockDim.x + threadIdx.x;
    const int f = (int)(idx & (OUTF - 1));
    const float invN  = 1.0f / (float)NPTS;
    const float mean  = stats[f] * invN;
    const float var   = stats[OUTF + f] * invN - mean * mean;
    const float scale = rsqrtf(var + 1e-6f) * gamma[f];
    float v = (x[idx] - mean) * scale + beta[f];
    x[idx] = (v >= 0.0f) ? v : 0.1f * v;
}

extern "C" void kernel_launch(void* const* d_in, const int* in_sizes, int n_in,
                              void* d_out, int out_size, void* d_ws, size_t ws_size,
                              hipStream_t stream)
{
    const float* qp   = (const float*)d_in[0];  // query_points   [65536,3]
    const float* sp   = (const float*)d_in[1];  // support_points [65536,3]
    const int*   nbr  = (const int*)  d_in[2];  // neighbors      [65536,32]
    const float* feat = (const float*)d_in[3];  // features       [65536,64]
    const float* kp   = (const float*)d_in[4];  // kernel_points  [15,3]
    const float* Wg   = (const float*)d_in[5];  // W              [15,64,128]
    const float* gam  = (const float*)d_in[6];  // gamma [128]
    const float* bet  = (const float*)d_in[7];  // beta  [128]
    float* out = (float*)d_out;                 // [65536,128]

    float* part  = (float*)d_ws;                // [2*64][128] partial sums
    float* stats = part + 2 * 64 * OUTF;        // [2][128] final sums

    // dynamic LDS: wbuf + 4*nx + 4*wl
    const size_t smem_bytes =
        (size_t)(16 * 16 * WBP + 4 * NNEI * NXP + 4 * 16 * WLP) * sizeof(float);

    kpconv_wmma<<<NPTS / 16, 128, smem_bytes, stream>>>(qp, sp, nbr, feat, kp, Wg, out);
    stats_partial<<<64, 128, 0, stream>>>(out, part);
    stats_final<<<1, 128, 0, stream>>>(part, stats);
    bn_leaky<<<(NPTS * OUTF) / 256, 256, 0, stream>>>(out, stats, gam, bet);
}